// GMM_3521873182881
// MI455X (gfx1250) — compile-verified
//
#include <hip/hip_runtime.h>
#include <math.h>

// GMM log-prob:  out[b] = logsumexp_k( alpha[k] - 0.5 * G[b,k] )
//   G = A @ W,  A = [x^2 | x] (2048 x 256),  W = [inv_var ; -2*m*inv_var] (256 x 512)
//   alpha[k] = log_softmax(log_weight)[k] - 0.5*(D*log(2pi) + sum_d(log_std + m^2*inv_var))
// W is stored pair-interleaved (fragment-ready for the 4x16 WMMA B layout):
//   Wf[pr][n] = { W[2*pr][n], W[2*pr+1][n] },  pr in [0,128), n in [0,512)

typedef __attribute__((ext_vector_type(2))) float v2f;
typedef __attribute__((ext_vector_type(8))) float v8f;
typedef __attribute__((address_space(3))) float lds_f32;

#define GMM_B 2048
#define GMM_K 512
#define GMM_D 128
#define GMM_KK 256           // inner GEMM dimension = 2*D
#define LOG2PI 1.8378770664093453f

// ---- CDNA5 async Global->LDS copy (ASYNCcnt-tracked) ----------------------
__device__ __forceinline__ void async_copy_b128(unsigned lds_byte,
                                                const float* gsrc) {
  asm volatile("global_load_async_to_lds_b128 %0, %1, off"
               :
               : "v"(lds_byte), "v"(gsrc)
               : "memory");
}

__device__ __forceinline__ void wait_asynccnt0() {
#if __has_builtin(__builtin_amdgcn_s_wait_asynccnt)
  __builtin_amdgcn_s_wait_asynccnt(0);
#else
  asm volatile("s_wait_asynccnt 0x0" ::: "memory");
#endif
}

// ---------------------------------------------------------------------------
// Prep 1: alpha[k] = (log_weight[k] - lse(log_weight)) - 0.5*(D*log2pi + c_k)
// ---------------------------------------------------------------------------
__global__ __launch_bounds__(512) void gmm_prep_alpha(
    const float* __restrict__ means, const float* __restrict__ log_std,
    const float* __restrict__ log_weight, float* __restrict__ alpha) {
  __shared__ float red[GMM_K];
  const int k = threadIdx.x;
  const float lw = log_weight[k];

  red[k] = lw;
  __syncthreads();
  for (int s = GMM_K / 2; s > 0; s >>= 1) {
    if (k < s) red[k] = fmaxf(red[k], red[k + s]);
    __syncthreads();
  }
  const float mxw = red[0];
  __syncthreads();
  red[k] = expf(lw - mxw);
  __syncthreads();
  for (int s = GMM_K / 2; s > 0; s >>= 1) {
    if (k < s) red[k] += red[k + s];
    __syncthreads();
  }
  const float lsew = mxw + logf(red[0]);

  float c = 0.0f;
  for (int d = 0; d < GMM_D; ++d) {
    const float ls = log_std[k * GMM_D + d];
    const float iv = expf(-ls);
    const float mm = means[k * GMM_D + d];
    c = fmaf(mm * mm, iv, c + ls);
  }
  alpha[k] = (lw - lsew) - 0.5f * ((float)GMM_D * LOG2PI + c);
}

// ---------------------------------------------------------------------------
// Prep 2: pair-interleaved W.  One float2 per thread, coalesced writes.
//   row 2*pr (< 128):  iv = exp(-log_std);  row >= 128:  -2*means*iv
//   (a pair never straddles the d=128 boundary)
// ---------------------------------------------------------------------------
__global__ __launch_bounds__(512) void gmm_prep_w(
    const float* __restrict__ means, const float* __restrict__ log_std,
    float2* __restrict__ Wf) {
  const int idx = blockIdx.x * blockDim.x + threadIdx.x;  // 0 .. 128*512-1
  const int pr = idx >> 9;
  const int k = idx & (GMM_K - 1);
  const int d0 = (pr * 2) & (GMM_D - 1);
  const float ls0 = log_std[k * GMM_D + d0];
  const float ls1 = log_std[k * GMM_D + d0 + 1];
  const float iv0 = expf(-ls0);
  const float iv1 = expf(-ls1);
  float2 o;
  if (pr < 64) {
    o.x = iv0;
    o.y = iv1;
  } else {
    o.x = -2.0f * means[k * GMM_D + d0] * iv0;
    o.y = -2.0f * means[k * GMM_D + d0 + 1] * iv1;
  }
  Wf[idx] = o;
}

// ---------------------------------------------------------------------------
// Main GEMM + partial logsumexp.
// Grid: (32 M-blocks of 64 rows, 8 N-blocks of 64 comps). Block: 128 thr (4 waves).
// Each wave: one 16(M) x 64(N) strip via 4 accumulators of
// V_WMMA_F32_16X16X4_F32 over the 256-deep inner dim, then 16-lane shuffle
// reductions to (rowmax, rowsum) partials.
// ---------------------------------------------------------------------------
__global__ __launch_bounds__(128) void gmm_gemm(
    const float* __restrict__ x, const v2f* __restrict__ Wf,
    const float* __restrict__ alpha, float2* __restrict__ partials) {
  // A tile: 64 rows x 256 cols fp32, XOR-swizzled (bits [5:2], 4-float
  // granularity) so the 16x4 A-fragment ds_load_b64s are bank-conflict free.
  __shared__ float At[64 * GMM_KK];

  const int tid = threadIdx.x;
  const int rowBase = blockIdx.x * 64;
  const int n0 = blockIdx.y * 64;

  // Stage A = [x^2 | x] in float4 granularity (swizzle preserves 16B groups):
  //   x^2 half: global_load_b128 + VALU squares + ds_store_b128
  //   x   half: global_load_async_to_lds_b128 (no VGPR round-trip)
#pragma unroll
  for (int it = 0; it < 16; ++it) {
    const int f4 = it * 128 + tid;   // float4 index within 64x128 x-tile
    const int r = f4 >> 5;           // tile row
    const int d = (f4 & 31) << 2;    // column (multiple of 4)
    const float* gp = x + (size_t)(rowBase + r) * GMM_D + d;
    const int swz = (r & 15) << 2;

    const unsigned lds_raw =
        (unsigned)(uintptr_t)(lds_f32*)&At[r * GMM_KK + ((GMM_D + d) ^ swz)];
    async_copy_b128(lds_raw, gp);

    const float4 v = *(const float4*)gp;
    *(float4*)&At[r * GMM_KK + (d ^ swz)] =
        make_float4(v.x * v.x, v.y * v.y, v.z * v.z, v.w * v.w);
  }
  wait_asynccnt0();
  __syncthreads();

  const int lane = tid & 31;
  const int wave = tid >> 5;
  const int half = lane >> 4;   // K-half of the 16x4 / 4x16 fragments
  const int nl = lane & 15;     // M index (A) / N index (B,C,D)
  const int arow = wave * 16 + nl;
  const int swA = nl << 2;

  v8f acc[4] = {};
  // B fragment base: inner-row-pair (2*ks + half), column (n0 + nl).
  const v2f* Wfp = Wf + (size_t)half * GMM_K + n0 + nl;

#pragma unroll 4
  for (int ks = 0; ks < GMM_KK / 4; ++ks) {
    const int kk = ks * 4;
    // A fragment: lane(half,m) holds A[m][kk + 2*half + {0,1}] (swizzled LDS)
    const v2f a = *(const v2f*)&At[arow * GMM_KK + ((kk + 2 * half) ^ swA)];
    const v2f* wrow = Wfp + (size_t)ks * 2 * GMM_K;
#pragma unroll
    for (int s = 0; s < 4; ++s) {
      const v2f b = wrow[s * 16];  // {W[kk+2h][n], W[kk+2h+1][n]} in one b64
      acc[s] = __builtin_amdgcn_wmma_f32_16x16x4_f32(
          false, a, false, b, (short)0, acc[s], false, false);
    }
  }

  // Epilogue: t = alpha[n] - 0.5*G; per-row (16-lane) max + sum(exp).
  const float al0 = alpha[n0 + nl];
  const float al1 = alpha[n0 + 16 + nl];
  const float al2 = alpha[n0 + 32 + nl];
  const float al3 = alpha[n0 + 48 + nl];

#pragma unroll
  for (int r = 0; r < 8; ++r) {
    const float t0 = al0 - 0.5f * acc[0][r];
    const float t1 = al1 - 0.5f * acc[1][r];
    const float t2 = al2 - 0.5f * acc[2][r];
    const float t3 = al3 - 0.5f * acc[3][r];
    float mx = fmaxf(fmaxf(t0, t1), fmaxf(t2, t3));
#pragma unroll
    for (int off = 1; off < 16; off <<= 1)
      mx = fmaxf(mx, __shfl_xor(mx, off, 32));
    float e = expf(t0 - mx) + expf(t1 - mx) + expf(t2 - mx) + expf(t3 - mx);
#pragma unroll
    for (int off = 1; off < 16; off <<= 1)
      e += __shfl_xor(e, off, 32);
    if (nl == 0) {
      const int grow = rowBase + wave * 16 + half * 8 + r;  // C/D row layout
      partials[(size_t)grow * 8 + blockIdx.y] = make_float2(mx, e);
    }
  }
}

// ---------------------------------------------------------------------------
// Combine the 8 N-block partial logsumexps per row.
// ---------------------------------------------------------------------------
__global__ __launch_bounds__(256) void gmm_combine(
    const float2* __restrict__ partials, float* __restrict__ out) {
  const int row = blockIdx.x * blockDim.x + threadIdx.x;
  if (row >= GMM_B) return;
  float2 p[8];
  float mx = -INFINITY;
#pragma unroll
  for (int i = 0; i < 8; ++i) {
    p[i] = partials[(size_t)row * 8 + i];
    mx = fmaxf(mx, p[i].x);
  }
  float s = 0.0f;
#pragma unroll
  for (int i = 0; i < 8; ++i) s += p[i].y * expf(p[i].x - mx);
  out[row] = mx + logf(s);
}

// ---------------------------------------------------------------------------
extern "C" void kernel_launch(void* const* d_in, const int* in_sizes, int n_in,
                              void* d_out, int out_size, void* d_ws,
                              size_t ws_size, hipStream_t stream) {
  const float* x = (const float*)d_in[0];
  const float* means = (const float*)d_in[1];
  const float* log_std = (const float*)d_in[2];
  const float* log_weight = (const float*)d_in[3];
  float* out = (float*)d_out;

  // Workspace layout (bytes): Wf 524288 | alpha 2048 | partials 131072
  char* ws = (char*)d_ws;
  float2* Wf = (float2*)ws;
  float* alpha = (float*)(ws + (size_t)GMM_KK * GMM_K * 4);
  float2* partials = (float2*)(ws + (size_t)GMM_KK * GMM_K * 4 + GMM_K * 4);

  gmm_prep_alpha<<<1, 512, 0, stream>>>(means, log_std, log_weight, alpha);
  gmm_prep_w<<<(GMM_KK / 2 * GMM_K) / 512, 512, 0, stream>>>(means, log_std,
                                                             Wf);
  gmm_gemm<<<dim3(GMM_B / 64, GMM_K / 64), 128, 0, stream>>>(
      x, (const v2f*)Wf, alpha, partials);
  gmm_combine<<<GMM_B / 256, 256, 0, stream>>>(partials, out);
}